// Node2VecModel_81973745812008
// MI455X (gfx1250) — compile-verified
//
#include <hip/hip_runtime.h>
#include <hip/hip_bf16.h>

typedef __attribute__((ext_vector_type(2))) float v2f;
typedef __attribute__((ext_vector_type(8))) float v8f;

#define NNODES 6
#define EMB    32
#define H1     64
#define H2     32

// ---------------------------------------------------------------------------
// C[16 x NC] = act( A[16 x KD] (LDS,row-major) @ B[KD x NC] (LDS,row-major) + bias )
// using V_WMMA_F32_16X16X4_F32 per ISA 7.12.2 fp32 layouts.
//   A frag (2 VGPR): lane L -> (M = L%16, K = reg + 2*(L/16))
//   B frag (2 VGPR): lane L -> (K = reg + 2*(L/16), N = L%16)   [transposed A layout]
//   C/D  (8 VGPR):   lane L, reg r -> (M = r + 8*(L/16), N = L%16)
// bias may be nullptr (global pointer, indexed by absolute column).
// ---------------------------------------------------------------------------
template <int KD, int NC, bool RELU>
__device__ __forceinline__ void gemm_tile(const float* __restrict__ sA,
                                          const float* __restrict__ sB,
                                          float* __restrict__ sC,
                                          const float* __restrict__ bias,
                                          int lane) {
  const int half = lane >> 4;   // 0 or 1
  const int l16  = lane & 15;
#pragma unroll
  for (int n0 = 0; n0 < NC; n0 += 16) {
    v8f acc = {};
#pragma unroll
    for (int k0 = 0; k0 < KD; k0 += 4) {
      const int ka = k0 + 2 * half;
      v2f a, b;
      a.x = sA[l16 * KD + ka + 0];
      a.y = sA[l16 * KD + ka + 1];
      b.x = sB[(ka + 0) * NC + n0 + l16];
      b.y = sB[(ka + 1) * NC + n0 + l16];
      // 8 args: (neg_a, A, neg_b, B, c_mod, C, reuse_a, reuse_b)
      acc = __builtin_amdgcn_wmma_f32_16x16x4_f32(false, a, false, b,
                                                  (short)0, acc, false, false);
    }
    const float bn = bias ? bias[n0 + l16] : 0.0f;
#pragma unroll
    for (int r = 0; r < 8; ++r) {
      const int row = r + 8 * half;
      float v = acc[r] + bn;
      if (RELU) v = fmaxf(v, 0.0f);
      sC[row * NC + n0 + l16] = v;
    }
  }
}

__global__ __launch_bounds__(32) void node2vec_fused_kernel(
    const float* __restrict__ x,      // [6,32]
    const int*   __restrict__ path,   // [7] (gather is computed-but-unused in ref)
    const float* __restrict__ fc1_w,  // [32,64]
    const float* __restrict__ fc1_b,  // [64]
    const float* __restrict__ fc2_w,  // [64,32]
    const float* __restrict__ fc2_b,  // [32]
    const float* __restrict__ fc3_w,  // [32,1]
    const float* __restrict__ fc3_b,  // [1]
    const float* __restrict__ pfc_w,  // [32,6]
    const float* __restrict__ pfc_b,  // [6]
    float* __restrict__ out)          // [13] = energy[6] ++ path_indices[7]
{
  __shared__ __align__(16) float sX [16 * EMB];     // padded x, 16x32
  __shared__ __align__(16) float sB [EMB * H1];     // weight staging (max 32x64)
  __shared__ __align__(16) float sH1[16 * H1];      // 16x64
  __shared__ __align__(16) float sH2[16 * H2];      // 16x32
  __shared__ __align__(16) float sLG[16 * 16];      // padded logits tile
  __shared__ float sLP[NNODES * NNODES];            // log-softmax rows

  const int lane = threadIdx.x;                     // single wave32

  // ---- stage x (zero-pad rows 6..15) : 512 floats, uniform 16 iters ----
  for (int i = lane; i < 16 * EMB; i += 32) {
    const int r = i / EMB, c = i % EMB;
    sX[i] = (r < NNODES) ? x[r * EMB + c] : 0.0f;
  }
  // ---- stage fc1_w (2048 floats as 512 float4, uniform 16 iters) ----
  for (int i = lane; i < (EMB * H1) / 4; i += 32)
    ((float4*)sB)[i] = ((const float4*)fc1_w)[i];
  __syncthreads();

  // ---- fc1 + relu : [16x32]@[32x64] -> sH1 ----
  gemm_tile<EMB, H1, true>(sX, sB, sH1, fc1_b, lane);
  __syncthreads();

  // ---- stage fc2_w [64x32] ----
  for (int i = lane; i < (H1 * H2) / 4; i += 32)
    ((float4*)sB)[i] = ((const float4*)fc2_w)[i];
  __syncthreads();

  // ---- fc2 + relu : [16x64]@[64x32] -> sH2 ----
  gemm_tile<H1, H2, true>(sH1, sB, sH2, fc2_b, lane);
  __syncthreads();

  // ---- stage path_fc_w [32,6] zero-padded to [32,16] (512, uniform) ----
  for (int i = lane; i < EMB * 16; i += 32) {
    const int k = i / 16, c = i % 16;
    sB[i] = (c < NNODES) ? pfc_w[k * NNODES + c] : 0.0f;
  }
  __syncthreads();

  // ---- path_fc logits : [16x32]@[32x16] -> sLG (bias added later) ----
  gemm_tile<EMB, 16, false>(sX, sB, sLG, nullptr, lane);
  __syncthreads();

  // ---- fc3 + softplus (per-lane dot, lanes 0..5), energy -> out[0..5] ----
  if (lane < NNODES) {
    float z = fc3_b[0];
#pragma unroll
    for (int k = 0; k < H2; ++k) z += sH2[lane * H2 + k] * fc3_w[k];
    // softplus = logaddexp(z, 0), numerically stable
    out[lane] = fmaxf(z, 0.0f) + log1pf(expf(-fabsf(z)));
  }

  // ---- row-wise log_softmax over 6 valid columns (lanes 0..5) ----
  if (lane < NNODES) {
    float lg[NNODES], m = -3.402823466e38f;
#pragma unroll
    for (int c = 0; c < NNODES; ++c) {
      lg[c] = sLG[lane * 16 + c] + pfc_b[c];
      m = fmaxf(m, lg[c]);
    }
    float s = 0.0f;
#pragma unroll
    for (int c = 0; c < NNODES; ++c) s += expf(lg[c] - m);
    const float lse = m + logf(s);
#pragma unroll
    for (int c = 0; c < NNODES; ++c) sLP[lane * NNODES + c] = lg[c] - lse;
  }
  __syncthreads();

  // ---- sequential multinomial walk (inherently serial -> lane 0) ----
  // Deterministic splitmix64 stream stands in for JAX threefry (not
  // bit-reproducible off-device); inverse-CDF draw over unvisited nodes.
  if (lane == 0) {
    (void)path;                       // reference gathers x[path] but discards it
    bool visited[NNODES] = {true, false, false, false, false, false};
    int cur = 0;
    unsigned long long seed = 42ull * 0x9E3779B97F4A7C15ull;
    out[NNODES] = 0.0f;               // path starts at node 0
    for (int step = 0; step < NNODES - 1; ++step) {
      float p[NNODES], tot = 0.0f;
      for (int j = 0; j < NNODES; ++j) {
        p[j] = visited[j] ? 0.0f : expf(sLP[cur * NNODES + j]);
        tot += p[j];
      }
      unsigned long long zv = (seed += 0x9E3779B97F4A7C15ull);
      zv = (zv ^ (zv >> 30)) * 0xBF58476D1CE4E5B9ull;
      zv = (zv ^ (zv >> 27)) * 0x94D049BB133111EBull;
      zv ^= (zv >> 31);
      const float u = (float)(zv >> 40) * (1.0f / 16777216.0f);   // [0,1)
      const float thr = u * tot;
      float csum = 0.0f;
      int nxt = -1;
      for (int j = 0; j < NNODES; ++j) {
        if (!visited[j]) {
          csum += p[j];
          if (nxt < 0 && thr < csum) nxt = j;
        }
      }
      if (nxt < 0)                    // numeric guard: take last unvisited
        for (int j = NNODES - 1; j >= 0; --j)
          if (!visited[j]) { nxt = j; break; }
      visited[nxt] = true;
      cur = nxt;
      out[NNODES + 1 + step] = (float)nxt;
    }
    out[2 * NNODES] = 0.0f;           // path ends at node 0
  }
}

extern "C" void kernel_launch(void* const* d_in, const int* in_sizes, int n_in,
                              void* d_out, int out_size, void* d_ws, size_t ws_size,
                              hipStream_t stream) {
  (void)in_sizes; (void)n_in; (void)out_size; (void)d_ws; (void)ws_size;
  const float* x     = (const float*)d_in[0];
  const int*   path  = (const int*)  d_in[1];
  const float* fc1_w = (const float*)d_in[2];
  const float* fc1_b = (const float*)d_in[3];
  const float* fc2_w = (const float*)d_in[4];
  const float* fc2_b = (const float*)d_in[5];
  const float* fc3_w = (const float*)d_in[6];
  const float* fc3_b = (const float*)d_in[7];
  const float* pfc_w = (const float*)d_in[8];
  const float* pfc_b = (const float*)d_in[9];

  // Latency-bound problem: one fused kernel, one wave32 workgroup.
  node2vec_fused_kernel<<<1, 32, 0, stream>>>(
      x, path, fc1_w, fc1_b, fc2_w, fc2_b, fc3_w, fc3_b, pfc_w, pfc_b,
      (float*)d_out);
}